// CategoricalActivation_51256139710941
// MI455X (gfx1250) — compile-verified
//
#include <hip/hip_runtime.h>

// CategoricalActivation for MI455X (gfx1250, wave32).
// Elementwise op: 268 MB stream -> ~11.5 us floor @ 23.3 TB/s; ~35 VALU/elem
// (IEEE-exact softsign divide + 7-way bucket count) makes it borderline
// VALU-bound. Data path uses CDNA5 async global->LDS loads (ASYNCcnt) in a
// double-buffered pipeline: register-free deep prefetch overlapping the
// divide-heavy compute; consume via ds_load_b128; NT stores for the output.

typedef float v4f __attribute__((ext_vector_type(4)));

#define T_DIM   1024
#define B_DIM   64
#define H_DIM   512
#define NCLS    8
#define BHD     (B_DIM * H_DIM)       // 32768 columns
#define TSTEP   8                     // t-rows per LDS tile (2 KB/row)
#define NTILES  8                     // tiles per block
#define TCHUNK  (TSTEP * NTILES)      // 64 t-steps per block
#define TILE_BYTES (TSTEP * H_DIM * 4)  // 16 KB per buffer

__device__ __forceinline__ float softsign_f(float v) {
  // IEEE round-to-nearest divide: bit-exact vs. jnp x/(1+|x|)
  return v / (1.0f + __builtin_fabsf(v));
}

// Issue one tile (TSTEP rows x 16B/lane) of async global->LDS b128 copies.
// Per-lane LDS target == per-lane future read location -> no barrier needed,
// only s_wait_asynccnt (in-order completion).
__device__ __forceinline__ void issue_async_tile(const float* gbase,
                                                 unsigned lds_off) {
#pragma unroll
  for (int ts = 0; ts < TSTEP; ++ts) {
    asm volatile("global_load_async_to_lds_b128 %0, %1, off"
                 :
                 : "v"(lds_off + (unsigned)(ts * H_DIM * 4)),
                   "v"(gbase + (size_t)ts * BHD)
                 : "memory");
  }
}

// Pass 1: boundaries[k,b,h] = softsign(x[boundary_idx[k,b,h], b, h]).
// 229376 scattered 4B gathers done exactly once; 896 KB result stays in L2.
__global__ __launch_bounds__(256)
void bnd_precompute_kernel(const float* __restrict__ x,
                           const int*   __restrict__ bidx,
                           float*       __restrict__ bnd,
                           int n) {
  int i = blockIdx.x * blockDim.x + threadIdx.x;
  if (i >= n) return;
  int col = i % BHD;                       // b*H + h
  int t   = bidx[i];                       // in [0, T)
  float v = x[(size_t)t * BHD + col];
  bnd[i] = softsign_f(v);
}

// Pass 2: streaming transform. Block = (b, all of H, one 64-step t-chunk).
// 128 threads x float4 covers H=512; async double-buffer through LDS.
__global__ __launch_bounds__(128)
void categorical_main_kernel(const float* __restrict__ x,
                             const float* __restrict__ crand,
                             const float* __restrict__ orand,
                             const float* __restrict__ rcls,
                             const int*   __restrict__ ncp,
                             const float* __restrict__ bnd,   // may be null
                             const int*   __restrict__ bidx,  // fallback path
                             float*       __restrict__ out) {
  __shared__ __align__(16) float lds_tile[2 * TSTEP * H_DIM];  // 32 KB

  const int b   = blockIdx.x % B_DIM;
  const int tc  = blockIdx.x / B_DIM;
  const int t0  = tc * TCHUNK;
  const int tid = threadIdx.x;
  const int h   = tid << 2;                 // 4 h-columns per thread
  const int col = b * H_DIM + h;

  // LDS byte offset of this thread's 16B slot within the wave's LDS space
  unsigned lds_base =
      (unsigned)(unsigned long long)(__attribute__((address_space(3))) float*)lds_tile;
  unsigned my_lds = lds_base + (unsigned)(tid * 16);

  const float* gcol = x + (size_t)t0 * BHD + col;

  // Kick off tile 0 into buffer 0 before doing the per-column setup, so the
  // first HBM transfer overlaps the mask/boundary loads below.
  issue_async_tile(gcol, my_lds);

  // ---- per-column state, hoisted out of the t loop ----
  v4f cr  = *(const v4f*)(crand + col);
  v4f orr = *(const v4f*)(orand + col);
  const float rc0 = rcls[0], rc1 = rcls[1], rc2 = rcls[2], rc3 = rcls[3];
  const float halfnc = 0.5f * (float)(*ncp);   // 4.0 for NC=8

  v4f bv[NCLS - 1];
  if (bnd != nullptr) {
#pragma unroll
    for (int k = 0; k < NCLS - 1; ++k)
      bv[k] = *(const v4f*)(bnd + (size_t)k * BHD + col);
  } else {
#pragma unroll
    for (int k = 0; k < NCLS - 1; ++k) {
#pragma unroll
      for (int j = 0; j < 4; ++j) {
        int t = bidx[(size_t)k * BHD + col + j];
        bv[k][j] = softsign_f(x[(size_t)t * BHD + col + j]);
      }
    }
  }

  bool catm[4], ordm[4];
#pragma unroll
  for (int j = 0; j < 4; ++j) {
    catm[j] = cr[j]  < 0.1f;
    ordm[j] = (orr[j] < 0.7f) && catm[j];
  }

  // ---- double-buffered pipeline over NTILES tiles ----
  for (int tile = 0; tile < NTILES; ++tile) {
    const int p = tile & 1;

    if (tile + 1 < NTILES) {
      // issue next tile into the other buffer, then wait for current tile:
      // completion is in-order, so asynccnt<=TSTEP means tile `tile` landed.
      issue_async_tile(gcol + (size_t)(tile + 1) * TSTEP * BHD,
                       my_lds + (unsigned)((p ^ 1) * TILE_BYTES));
      asm volatile("s_wait_asynccnt 0x8" ::: "memory");
    } else {
      asm volatile("s_wait_asynccnt 0x0" ::: "memory");
    }

    size_t obase = (size_t)(t0 + tile * TSTEP) * BHD + col;
#pragma unroll
    for (int ts = 0; ts < TSTEP; ++ts) {
      v4f xin = *(const v4f*)&lds_tile[(p * TSTEP + ts) * H_DIM + h];  // ds_load_b128
      v4f res;
#pragma unroll
      for (int j = 0; j < 4; ++j) {
        float sv = softsign_f(xin[j]);     // rounded value used for BOTH
        int cnt = 0;                       // compare and output (bit-exact)
#pragma unroll
        for (int k = 0; k < NCLS - 1; ++k)
          cnt += (sv > bv[k][j]) ? 1 : 0;  // v_cmp + carry-add

        float xc = (float)cnt - halfnc;    // counts - NC/2, exact integer fp

        // relabel: xc==c only possible for c = cnt-4 in [0,3] -> branchless
        // 3-level cndmask tree on cnt's bits (cnt in [0,7])
        float lo01 = (cnt & 1) ? rc1 : rc0;
        float lo23 = (cnt & 1) ? rc3 : rc2;
        float sel  = (cnt & 2) ? lo23 : lo01;
        float xr   = (cnt & 4) ? sel  : 0.0f;

        float cv = ordm[j] ? xr : xc;
        res[j]   = catm[j] ? cv : sv;
      }
      __builtin_nontemporal_store(res, (v4f*)(out + obase + (size_t)ts * BHD));
    }
  }
}

extern "C" void kernel_launch(void* const* d_in, const int* in_sizes, int n_in,
                              void* d_out, int out_size, void* d_ws, size_t ws_size,
                              hipStream_t stream) {
  const float* x     = (const float*)d_in[0];   // [T,B,H] f32
  const float* crand = (const float*)d_in[1];   // [B,H]   f32
  const float* orand = (const float*)d_in[2];   // [B,H]   f32
  const float* rcls  = (const float*)d_in[3];   // [NC]    f32
  const int*   bidx  = (const int*)d_in[4];     // [NC-1,B,H] i32
  const int*   ncp   = (const int*)d_in[5];     // scalar NC
  float* out = (float*)d_out;

  (void)in_sizes; (void)n_in; (void)out_size;

  const int nb = (NCLS - 1) * BHD;              // 229376 boundary values
  float* bnd = nullptr;
  if (ws_size >= (size_t)nb * sizeof(float)) {
    bnd = (float*)d_ws;
    bnd_precompute_kernel<<<(nb + 255) / 256, 256, 0, stream>>>(x, bidx, bnd, nb);
  }

  dim3 grid(B_DIM * (T_DIM / TCHUNK));          // 1024 blocks x 4 wave32
  categorical_main_kernel<<<grid, H_DIM / 4, 0, stream>>>(
      x, crand, orand, rcls, ncp, bnd, bidx, out);
}